// Attention_23003844837848
// MI455X (gfx1250) — compile-verified
//
#include <hip/hip_runtime.h>
#include <hip/hip_bf16.h>
#include <math.h>

#define H_DIM 1024
#define B_DIM 32
#define S_DIM 8192

typedef float v2f __attribute__((ext_vector_type(2)));
typedef float v8f __attribute__((ext_vector_type(8)));

// ---------------------------------------------------------------------------
// K1: q[32,1024] = dec[32,1024] @ W[1024,1024]   (q[m,n] = sum_k dec[m,k]*W[k,n])
// One wave per 16x16 C tile, K stepped by 4 via V_WMMA_F32_16X16X4_F32.
// A 16x4 layout : lane<16 -> (m=lane, K=0/1 in v0/v1); lane>=16 -> (m=lane-16, K=2/3)
// B 4x16 layout : v0: lanes0-15=K0, lanes16-31=K2 ; v1: lanes0-15=K1, lanes16-31=K3
// C 16x16 layout: VGPR r: lanes0-15 row m=r, lanes16-31 row m=r+8, col = lane&15
// ---------------------------------------------------------------------------
__global__ void qproj_wmma_kernel(const float* __restrict__ dec,
                                  const float* __restrict__ W,
                                  float* __restrict__ q) {
    const int lane  = threadIdx.x & 31;
    const int wave  = (blockIdx.x * blockDim.x + threadIdx.x) >> 5; // 0..127
    const int mtile = wave & 1;        // 2 tiles of M (B_DIM=32)
    const int ntile = wave >> 1;       // 64 tiles of N (H_DIM=1024)

    const int mrow  = lane & 15;
    const int khalf = lane >> 4;       // 0 or 1
    const int ncol  = ntile * 16 + (lane & 15);

    const float* __restrict__ arow = dec + (mtile * 16 + mrow) * H_DIM + khalf * 2;
    const float* __restrict__ bcol = W + (size_t)(khalf * 2) * H_DIM + ncol;

    v8f c = {};
    #pragma unroll 4
    for (int k0 = 0; k0 < H_DIM; k0 += 4) {
        v2f a, b;
        a.x = arow[k0 + 0];                     // dec[m, k0 + 2*khalf + 0]
        a.y = arow[k0 + 1];                     // dec[m, k0 + 2*khalf + 1]
        b.x = bcol[(size_t)k0 * H_DIM];         // W[k0 + 2*khalf + 0, n]
        b.y = bcol[(size_t)k0 * H_DIM + H_DIM]; // W[k0 + 2*khalf + 1, n]
        c = __builtin_amdgcn_wmma_f32_16x16x4_f32(
                /*neg_a=*/false, a, /*neg_b=*/false, b,
                /*c_mod=*/(short)0, c, /*reuse_a=*/false, /*reuse_b=*/false);
    }

    const int mhi = khalf * 8;
    #pragma unroll
    for (int r = 0; r < 8; ++r) {
        q[(size_t)(mtile * 16 + mhi + r) * H_DIM + ntile * 16 + (lane & 15)] = c[r];
    }
}

// ---------------------------------------------------------------------------
// K2: energies[b,s] = sum_h q[b,h] * enc[s,b,h]   -> out[b*S + s]
// Grid (S/64, B). Block = 256 threads = 8 waves; each wave computes 8 s-values.
// Lane l covers h = l*4 + 128*j (j=0..7): fully coalesced float4 streams.
// This is THE bandwidth kernel (reads enc exactly once = 1 GiB).
// ---------------------------------------------------------------------------
__global__ void energies_kernel(const float* __restrict__ enc,
                                const float* __restrict__ q,
                                float* __restrict__ out) {
    const int b    = blockIdx.y;
    const int lane = threadIdx.x & 31;
    const int wave = threadIdx.x >> 5;                 // 0..7

    // Per-lane slice of q[b,:] held in registers (8 x float4 = 32 VGPRs)
    float4 qr[8];
    #pragma unroll
    for (int j = 0; j < 8; ++j)
        qr[j] = *reinterpret_cast<const float4*>(q + (size_t)b * H_DIM + lane * 4 + j * 128);

    const int s_base = blockIdx.x * 64 + wave * 8;
    #pragma unroll 2
    for (int i = 0; i < 8; ++i) {
        const int s = s_base + i;
        const float* __restrict__ ep =
            enc + ((size_t)s * B_DIM + b) * H_DIM + lane * 4;
        float acc = 0.0f;
        #pragma unroll
        for (int j = 0; j < 8; ++j) {
            float4 e = *reinterpret_cast<const float4*>(ep + j * 128);
            acc = fmaf(e.x, qr[j].x, acc);
            acc = fmaf(e.y, qr[j].y, acc);
            acc = fmaf(e.z, qr[j].z, acc);
            acc = fmaf(e.w, qr[j].w, acc);
        }
        // wave32 butterfly reduction
        #pragma unroll
        for (int off = 16; off > 0; off >>= 1)
            acc += __shfl_xor(acc, off, 32);
        if (lane == 0)
            out[(size_t)b * S_DIM + s] = acc;
    }
}

// ---------------------------------------------------------------------------
// K3: in-place softmax over each row of out[B,S]. One block per row.
// 256 threads x 32 elements each, held in registers between passes.
// (Bias term dec[b].bias omitted: softmax is shift-invariant per row.)
// ---------------------------------------------------------------------------
__global__ void softmax_kernel(float* __restrict__ out) {
    const int b = blockIdx.x;
    const int t = threadIdx.x;     // 0..255
    float* __restrict__ row = out + (size_t)b * S_DIM;

    float v[32];
    float m = -INFINITY;
    #pragma unroll
    for (int i = 0; i < 32; ++i) {
        v[i] = row[t + i * 256];
        m = fmaxf(m, v[i]);
    }

    __shared__ float red[256];
    red[t] = m;
    __syncthreads();
    for (int off = 128; off > 0; off >>= 1) {
        if (t < off) red[t] = fmaxf(red[t], red[t + off]);
        __syncthreads();
    }
    m = red[0];
    __syncthreads();

    float sum = 0.0f;
    #pragma unroll
    for (int i = 0; i < 32; ++i) {
        v[i] = expf(v[i] - m);
        sum += v[i];
    }
    red[t] = sum;
    __syncthreads();
    for (int off = 128; off > 0; off >>= 1) {
        if (t < off) red[t] += red[t + off];
        __syncthreads();
    }
    const float inv = 1.0f / red[0];
    #pragma unroll
    for (int i = 0; i < 32; ++i)
        row[t + i * 256] = v[i] * inv;
}

// ---------------------------------------------------------------------------
extern "C" void kernel_launch(void* const* d_in, const int* in_sizes, int n_in,
                              void* d_out, int out_size, void* d_ws, size_t ws_size,
                              hipStream_t stream) {
    const float* dec = (const float*)d_in[0];   // [B, H]
    const float* enc = (const float*)d_in[1];   // [S, B, H]
    const float* W   = (const float*)d_in[2];   // [H, H]
    // d_in[3] (bias) unused: softmax is invariant to per-row constants.
    float* out = (float*)d_out;                 // [B, S]
    float* q   = (float*)d_ws;                  // [B, H] = 128 KB scratch

    // K1: q = dec @ W  (128 waves = 32 blocks x 128 threads)
    qproj_wmma_kernel<<<32, 128, 0, stream>>>(dec, W, q);

    // K2: energies -> d_out  (4096 blocks, streams enc once: ~1 GiB)
    energies_kernel<<<dim3(S_DIM / 64, B_DIM), 256, 0, stream>>>(enc, q, out);

    // K3: row softmax in place
    softmax_kernel<<<B_DIM, 256, 0, stream>>>(out);
}